// PointNet_SA_AT_Module_55121610277163
// MI455X (gfx1250) — compile-verified
//
#include <hip/hip_runtime.h>
#include <hip/hip_bf16.h>
#include <math.h>

#define BB   8
#define NN   4096
#define CFF  64
#define MM   1024
#define KK   32
#define PT   262144          // B*M*K positions
#define RAD2 0.04f           // radius^2
#define EPS_ 1e-5f

typedef float v2f __attribute__((ext_vector_type(2)));
typedef float v8f __attribute__((ext_vector_type(8)));

// ---------------------------------------------------------------------------
// 1) Farthest point sampling: one block per batch, 1024 threads, dists in regs
// ---------------------------------------------------------------------------
__global__ __launch_bounds__(1024) void fps_kernel(const float* __restrict__ xyz,
                                                   float* __restrict__ new_xyz_out) {
    const int b   = blockIdx.x;
    const int tid = threadIdx.x;
    const float* xb = xyz + (size_t)b * NN * 3;

    float px[4], py[4], pz[4], dd[4];
#pragma unroll
    for (int i = 0; i < 4; ++i) {
        int j = tid + i * 1024;
        px[i] = xb[j * 3 + 0];
        py[i] = xb[j * 3 + 1];
        pz[i] = xb[j * 3 + 2];
        dd[i] = 1e10f;
    }

    __shared__ float cen[3];
    __shared__ float wd[32];
    __shared__ int   wi[32];
    __shared__ int   sfar;

    int far = 0;
    const int lane = tid & 31, wv = tid >> 5;
    for (int it = 0; it < MM; ++it) {
        if (tid == 0) {
            float cx = xb[far * 3 + 0], cy = xb[far * 3 + 1], cz = xb[far * 3 + 2];
            cen[0] = cx; cen[1] = cy; cen[2] = cz;
            float* o = new_xyz_out + ((size_t)b * MM + it) * 3;
            o[0] = cx; o[1] = cy; o[2] = cz;
        }
        __syncthreads();
        float cx = cen[0], cy = cen[1], cz = cen[2];
        float bd = -1.0f; int bi = 0x7fffffff;
#pragma unroll
        for (int i = 0; i < 4; ++i) {
            float dx = px[i] - cx, dy = py[i] - cy, dz = pz[i] - cz;
            float d = dx * dx + dy * dy + dz * dz;
            dd[i] = fminf(dd[i], d);
            int j = tid + i * 1024;
            if (dd[i] > bd || (dd[i] == bd && j < bi)) { bd = dd[i]; bi = j; }
        }
        // wave32 argmax (min index on tie -> matches jnp.argmax first-occurrence)
        for (int off = 16; off >= 1; off >>= 1) {
            float od = __shfl_xor(bd, off);
            int   oi = __shfl_xor(bi, off);
            if (od > bd || (od == bd && oi < bi)) { bd = od; bi = oi; }
        }
        if (lane == 0) { wd[wv] = bd; wi[wv] = bi; }
        __syncthreads();
        if (wv == 0) {
            bd = wd[lane]; bi = wi[lane];
            for (int off = 16; off >= 1; off >>= 1) {
                float od = __shfl_xor(bd, off);
                int   oi = __shfl_xor(bi, off);
                if (od > bd || (od == bd && oi < bi)) { bd = od; bi = oi; }
            }
            if (lane == 0) sfar = bi;
        }
        __syncthreads();
        far = sfar;
    }
}

// ---------------------------------------------------------------------------
// 2) Ball query + gather + concat -> act0[68][PT]; one wave per (b,m) query.
// ---------------------------------------------------------------------------
__global__ __launch_bounds__(256) void group_kernel(const float* __restrict__ xyz,
                                                    const float* __restrict__ points,
                                                    const float* __restrict__ new_xyz,
                                                    float* __restrict__ act0) {
    int wid  = blockIdx.x * 8 + (threadIdx.x >> 5);
    int lane = threadIdx.x & 31;
    int b = wid / MM, m = wid % MM;
    const float* xb = xyz + (size_t)b * NN * 3;
    const float* nx = new_xyz + ((size_t)b * MM + m) * 3;
    float cx = nx[0], cy = nx[1], cz = nx[2];

    int myidx = -1;
    int cnt = 0;
    for (int base = 0; base < NN; base += 32) {
        int j = base + lane;
        float dx = xb[j * 3 + 0] - cx, dy = xb[j * 3 + 1] - cy, dz = xb[j * 3 + 2] - cz;
        float d = dx * dx + dy * dy + dz * dz;
        unsigned mask = (unsigned)__ballot(d <= RAD2);
        int pc = __popc(mask);
        if (myidx < 0) {
            int want = lane - cnt;            // rank this lane still needs
            if (want >= 0 && want < pc) {
                int n = want, nth = 0;
                for (int t = 0; t < 32; ++t) {
                    if (mask & (1u << t)) { if (n == 0) { nth = t; break; } --n; }
                }
                myidx = base + nth;
            }
        }
        cnt += pc;
        if (cnt >= KK) break;                 // uniform across the wave
    }
    int first = __shfl(myidx, 0);             // centroid itself always in-ball
    if (myidx < 0) myidx = first;
    if (myidx < 0) myidx = 0;                 // paranoia guard

    size_t p = ((size_t)b * MM + m) * KK + lane;
    act0[(size_t)0 * PT + p] = xb[myidx * 3 + 0] - cx;
    act0[(size_t)1 * PT + p] = xb[myidx * 3 + 1] - cy;
    act0[(size_t)2 * PT + p] = xb[myidx * 3 + 2] - cz;
    const float* pf = points + ((size_t)b * NN + myidx) * CFF;
#pragma unroll 4
    for (int c = 0; c < CFF; ++c) act0[(size_t)(3 + c) * PT + p] = pf[c];
    act0[(size_t)67 * PT + p] = 0.0f;         // zero pad row -> Cin multiple of 4
}

// ---------------------------------------------------------------------------
// 3) GEMM via V_WMMA_F32_16X16X4_F32, templated so the K-loop fully unrolls.
//    A-fragments (weights) preloaded once into registers; B loads clause and
//    overlap the WMMA chain. Fused input BN+ReLU and fused output BN stats.
//    Block = 256 thr = 8 waves = 4 (Cout 16-tiles) x 2 (P 16-tiles), each wave
//    loops 4 P-subtiles -> block computes 64 channels x 128 positions.
// ---------------------------------------------------------------------------
template <int CIN_PAD, int CIN_W, bool HAS_BN>
__global__ __launch_bounds__(256) void gemm_wmma_kernel(
    const float* __restrict__ W,      // [Cout][CIN_W]
    const float* __restrict__ act,    // [CIN_PAD][PT]
    float* __restrict__ out,          // [Cout][PT]
    const float* __restrict__ scale,  // per-input-channel BN scale
    const float* __restrict__ shift,  // per-input-channel BN shift
    float* __restrict__ ssum, float* __restrict__ ssq) {
    constexpr int STEPS = CIN_PAD / 4;

    __shared__ float lsum[64];
    __shared__ float lsq[64];
    const int tid = threadIdx.x;
    if (tid < 64) { lsum[tid] = 0.0f; lsq[tid] = 0.0f; }
    __syncthreads();

    const int lane = tid & 31, wv = tid >> 5;
    const int otile = wv >> 1, pt = wv & 1;
    const int half = lane >> 4, lr = lane & 15;
    const int o0 = blockIdx.y * 64 + otile * 16;
    const int row = o0 + lr;

    // Preload all A fragments for this wave's 16 output rows (subtile-invariant)
    float aA0[STEPS], aA1[STEPS];
#pragma unroll
    for (int s = 0; s < STEPS; ++s) {
        const int ca = 4 * s + (half << 1);
        aA0[s] = (ca     < CIN_W) ? W[row * CIN_W + ca    ] : 0.0f;
        aA1[s] = (ca + 1 < CIN_W) ? W[row * CIN_W + ca + 1] : 0.0f;
    }

    float rs[8], rq[8];
#pragma unroll
    for (int r = 0; r < 8; ++r) { rs[r] = 0.0f; rq[r] = 0.0f; }

    for (int sub = 0; sub < 4; ++sub) {
        const int p0 = blockIdx.x * 128 + sub * 32 + pt * 16;
        // prefetch next subtile's B columns (global_prefetch_b8)
        if (sub < 3) {
#pragma unroll
            for (int s = 0; s < STEPS; s += 8)
                __builtin_prefetch(&act[(size_t)(4 * s + (half << 1)) * PT + p0 + 32 + lr], 0, 1);
        }
        v8f acc = {0.f, 0.f, 0.f, 0.f, 0.f, 0.f, 0.f, 0.f};
#pragma unroll
        for (int s = 0; s < STEPS; ++s) {
            const int ca = 4 * s + (half << 1);
            float b0 = act[(size_t)ca * PT + p0 + lr];
            float b1 = act[(size_t)(ca + 1) * PT + p0 + lr];
            if (HAS_BN) {
                b0 = fmaxf(b0 * scale[ca] + shift[ca], 0.0f);
                b1 = fmaxf(b1 * scale[ca + 1] + shift[ca + 1], 0.0f);
            }
            v2f a, bv;
            a[0] = aA0[s]; a[1] = aA1[s];
            bv[0] = b0;    bv[1] = b1;
            acc = __builtin_amdgcn_wmma_f32_16x16x4_f32(
                false, a, false, bv, (short)0, acc, false, false);
        }
#pragma unroll
        for (int r = 0; r < 8; ++r) {
            const int ch = o0 + r + (half << 3);
            out[(size_t)ch * PT + p0 + lr] = acc[r];
            rs[r] += acc[r];
            rq[r] += acc[r] * acc[r];
        }
    }
    // reduce across the 16 lanes of each half (xor masks 1..8 stay in-half)
#pragma unroll
    for (int r = 0; r < 8; ++r) {
        float s = rs[r], q = rq[r];
        for (int off = 8; off >= 1; off >>= 1) {
            s += __shfl_xor(s, off);
            q += __shfl_xor(q, off);
        }
        if (lr == 0) {
            const int lc = otile * 16 + r + (half << 3);
            atomicAdd(&lsum[lc], s);
            atomicAdd(&lsq[lc], q);
        }
    }
    __syncthreads();
    if (tid < 64) {
        const int ch = blockIdx.y * 64 + tid;
        atomicAdd(&ssum[ch], lsum[tid]);
        atomicAdd(&ssq[ch], lsq[tid]);
    }
}

// per-channel (mean,var,gamma,beta) -> (scale, shift)
__global__ void bnparam_kernel(const float* __restrict__ ssum, const float* __restrict__ ssq,
                               const float* __restrict__ g, const float* __restrict__ bb,
                               float* __restrict__ s, float* __restrict__ t) {
    const int c = threadIdx.x;
    const float n = (float)PT;
    float mu  = ssum[c] / n;
    float var = ssq[c] / n - mu * mu;
    float sc  = g[c] * rsqrtf(var + EPS_);
    s[c] = sc;
    t[c] = bb[c] - mu * sc;
}

__global__ void zero_kernel(float* __restrict__ p, int n) {
    int i = blockIdx.x * blockDim.x + threadIdx.x;
    if (i < n) p[i] = 0.0f;
}

// ---------------------------------------------------------------------------
// 4) scSE: per-(b,c) spatial mean of BN'd features
// ---------------------------------------------------------------------------
__global__ __launch_bounds__(256) void cse_mean_kernel(const float* __restrict__ D2,
                                                       const float* __restrict__ s2,
                                                       const float* __restrict__ t2,
                                                       float* __restrict__ y) {
    __shared__ float red[256];
    const int b = blockIdx.x >> 8;
    const int c = blockIdx.x & 255;
    const size_t base = (size_t)c * PT + (size_t)b * (MM * KK);
    float s = 0.0f;
    for (int i = threadIdx.x; i < MM * KK; i += 256) s += D2[base + i];
    red[threadIdx.x] = s;
    __syncthreads();
    for (int st = 128; st >= 1; st >>= 1) {
        if (threadIdx.x < st) red[threadIdx.x] += red[threadIdx.x + st];
        __syncthreads();
    }
    if (threadIdx.x == 0)
        y[b * 256 + c] = s2[c] * (red[0] / (float)(MM * KK)) + t2[c];
}

// sigmoid(relu(y @ w1^T) @ w2^T): one block per batch
__global__ __launch_bounds__(256) void cse_mlp_kernel(const float* __restrict__ y,
                                                      const float* __restrict__ w1,
                                                      const float* __restrict__ w2,
                                                      float* __restrict__ ysc) {
    __shared__ float yb[256];
    __shared__ float h[16];
    const int b = blockIdx.x, t = threadIdx.x;
    yb[t] = y[b * 256 + t];
    __syncthreads();
    if (t < 16) {
        float a = 0.0f;
        for (int c = 0; c < 256; ++c) a += yb[c] * w1[t * 256 + c];
        h[t] = fmaxf(a, 0.0f);
    }
    __syncthreads();
    float z = 0.0f;
#pragma unroll
    for (int j = 0; j < 16; ++j) z += h[j] * w2[t * 16 + j];
    ysc[b * 256 + t] = 1.0f / (1.0f + expf(-z));
}

// ---------------------------------------------------------------------------
// 5) final: x_bn * (y_cse + sigmoid(sse.x_bn)) -> relu -> max over K -> out
// ---------------------------------------------------------------------------
__global__ __launch_bounds__(256) void scse_final_kernel(const float* __restrict__ D2,
                                                         const float* __restrict__ s2,
                                                         const float* __restrict__ t2,
                                                         const float* __restrict__ ysc,
                                                         const float* __restrict__ ssew,
                                                         float* __restrict__ outp) {
    __shared__ float qp[8][32];
    __shared__ float qv[32];
    const int bm = blockIdx.x;
    const int b = bm / MM;
    const size_t pbase = (size_t)bm * KK;
    const int t = threadIdx.x;
    const int wv = t >> 5, k = t & 31;

    float part = 0.0f;
    for (int c = wv * 32; c < wv * 32 + 32; ++c) {
        float x = s2[c] * D2[(size_t)c * PT + pbase + k] + t2[c];
        part += ssew[c] * x;
    }
    qp[wv][k] = part;
    __syncthreads();
    if (t < 32) {
        float q = 0.0f;
#pragma unroll
        for (int w = 0; w < 8; ++w) q += qp[w][t];
        qv[t] = 1.0f / (1.0f + expf(-q));
    }
    __syncthreads();

    const int c = t;
    const float ys = ysc[b * 256 + c];
    const float sc = s2[c], sh = t2[c];
    const float* dp = D2 + (size_t)c * PT + pbase;
    float mx = -1e30f;
#pragma unroll
    for (int kk = 0; kk < KK; ++kk) {
        float x = sc * dp[kk] + sh;
        mx = fmaxf(mx, x * (ys + qv[kk]));
    }
    outp[(size_t)bm * 256 + c] = fmaxf(mx, 0.0f);  // max(relu(.)) == relu(max(.))
}

// ---------------------------------------------------------------------------
extern "C" void kernel_launch(void* const* d_in, const int* in_sizes, int n_in,
                              void* d_out, int out_size, void* d_ws, size_t ws_size,
                              hipStream_t stream) {
    const float* xyz    = (const float*)d_in[0];
    const float* points = (const float*)d_in[1];
    const float* w0 = (const float*)d_in[2];
    const float* g0 = (const float*)d_in[3];
    const float* b0 = (const float*)d_in[4];
    const float* w1 = (const float*)d_in[5];
    const float* g1 = (const float*)d_in[6];
    const float* b1 = (const float*)d_in[7];
    const float* w2 = (const float*)d_in[8];
    const float* g2 = (const float*)d_in[9];
    const float* b2 = (const float*)d_in[10];
    const float* cse_w1 = (const float*)d_in[11];
    const float* cse_w2 = (const float*)d_in[12];
    const float* sse_w  = (const float*)d_in[13];

    float* out      = (float*)d_out;
    float* new_xyz  = out;                                // (B,M,3)
    float* feat_out = out + (size_t)BB * MM * 3;          // (B,M,256)

    float* ws = (float*)d_ws;
    size_t off = 0;
    float* act0 = ws + off; off += (size_t)68  * PT;      // grouped features (padded)
    float* D0   = ws + off; off += (size_t)64  * PT;      // conv0 raw output
    float* D1   = ws + off; off += (size_t)128 * PT;      // conv1 raw output
    float* D2   = ws + off; off += (size_t)256 * PT;      // conv2 raw output
    float* stats = ws + off; off += 896;                  // sums / sumsq
    float *sum0 = stats,       *sq0 = stats + 64;
    float *sum1 = stats + 128, *sq1 = stats + 256;
    float *sum2 = stats + 384, *sq2 = stats + 640;
    float* bnp = ws + off; off += 896;                    // folded BN scale/shift
    float *s0 = bnp,       *t0 = bnp + 64;
    float *s1 = bnp + 128, *t1 = bnp + 256;
    float *s2 = bnp + 384, *t2 = bnp + 640;
    float* yv  = ws + off; off += (size_t)BB * 256;       // cSE means
    float* ysc = ws + off; off += (size_t)BB * 256;       // cSE gates

    fps_kernel<<<BB, 1024, 0, stream>>>(xyz, new_xyz);
    zero_kernel<<<4, 256, 0, stream>>>(stats, 896);
    group_kernel<<<BB * MM / 8, 256, 0, stream>>>(xyz, points, new_xyz, act0);

    gemm_wmma_kernel<68, 67, false><<<dim3(PT / 128, 1), 256, 0, stream>>>(
        w0, act0, D0, nullptr, nullptr, sum0, sq0);
    bnparam_kernel<<<1, 64, 0, stream>>>(sum0, sq0, g0, b0, s0, t0);

    gemm_wmma_kernel<64, 64, true><<<dim3(PT / 128, 2), 256, 0, stream>>>(
        w1, D0, D1, s0, t0, sum1, sq1);
    bnparam_kernel<<<1, 128, 0, stream>>>(sum1, sq1, g1, b1, s1, t1);

    gemm_wmma_kernel<128, 128, true><<<dim3(PT / 128, 4), 256, 0, stream>>>(
        w2, D1, D2, s1, t1, sum2, sq2);
    bnparam_kernel<<<1, 256, 0, stream>>>(sum2, sq2, g2, b2, s2, t2);

    cse_mean_kernel<<<BB * 256, 256, 0, stream>>>(D2, s2, t2, yv);
    cse_mlp_kernel<<<BB, 256, 0, stream>>>(yv, cse_w1, cse_w2, ysc);
    scse_final_kernel<<<BB * MM, 256, 0, stream>>>(D2, s2, t2, ysc, sse_w, feat_out);
}